// GraphSAGE_24215025615234
// MI455X (gfx1250) — compile-verified
//
#include <hip/hip_runtime.h>
#include <math.h>

#define N_NODES 50000
#define N_EDGES 800000
#define F_DIM   128          // F_IN == HIDDEN == 128
#define N_CLASSES 40

typedef __attribute__((ext_vector_type(2))) float v2f;
typedef __attribute__((ext_vector_type(8))) float v8f;

__device__ __forceinline__ void atomic_add_f32(float* p, float v) {
    __hip_atomic_fetch_add(p, v, __ATOMIC_RELAXED, __HIP_MEMORY_SCOPE_AGENT);
}

// --- degree: one thread per edge --------------------------------------------
__global__ void sage_deg_kernel(const int* __restrict__ dst,
                                float* __restrict__ deg, int E) {
    int e = blockIdx.x * blockDim.x + threadIdx.x;
    if (e < E) atomic_add_f32(&deg[dst[e]], 1.0f);
}

// --- scatter-add: one wave per edge, float4 per lane (128 feats / 32 lanes) -
__global__ void sage_scatter_kernel(const float* __restrict__ feat,
                                    const int* __restrict__ src,
                                    const int* __restrict__ dst,
                                    float* __restrict__ agg, int E) {
    int gid   = blockIdx.x * blockDim.x + threadIdx.x;
    int wave  = gid >> 5;
    int lane  = gid & 31;
    int nwave = (gridDim.x * blockDim.x) >> 5;
    for (int e = wave; e < E; e += nwave) {
        int s = src[e];
        int d = dst[e];
        const float4 v = *(const float4*)(feat + (size_t)s * F_DIM + lane * 4);
        float* o = agg + (size_t)d * F_DIM + lane * 4;
        atomic_add_f32(o + 0, v.x);
        atomic_add_f32(o + 1, v.y);
        atomic_add_f32(o + 2, v.z);
        atomic_add_f32(o + 3, v.w);
    }
}

// --- mean: agg[i][:] /= max(deg[i],1) ---------------------------------------
__global__ void sage_scale_kernel(float* __restrict__ agg,
                                  const float* __restrict__ deg, int n) {
    int i = blockIdx.x * blockDim.x + threadIdx.x;
    if (i < n) {
        float d = deg[i >> 7];                 // /128
        agg[i] *= 1.0f / fmaxf(d, 1.0f);
    }
}

// --- fused dual GEMM + bias + ReLU via V_WMMA_F32_16X16X4_F32 ----------------
// out[M,Nout] = relu(Aagg @ Wl + Aroot @ Wr + bias);  K = 128 fixed.
// One wave per 16x16 output tile. A 16x4 f32: lane<16 -> K=0,1 ; lane>=16 -> K=2,3
// (contiguous float2). B 4x16 f32: VGPR0 = rows K=0 / K=2, VGPR1 = rows K=1 / K=3,
// lane&15 = column. C/D: VGPR v -> row v + 8*(lane>>4), col = lane&15.
__global__ void sage_gemm_kernel(const float* __restrict__ Aagg,
                                 const float* __restrict__ Aroot,
                                 const float* __restrict__ Wl,   // [128, Nout]
                                 const float* __restrict__ Wr,   // [128, Nout]
                                 const float* __restrict__ bias, // [Nout]
                                 float* __restrict__ out,        // [M, Nout]
                                 int Mtiles, int Ntiles, int Nout) {
    const int waveInBlk = threadIdx.x >> 5;
    const int lane      = threadIdx.x & 31;
    const int tile      = blockIdx.x * (blockDim.x >> 5) + waveInBlk;
    if (tile >= Mtiles * Ntiles) return;     // wave-uniform: EXEC all-1 for WMMA

    const int mt   = tile % Mtiles;          // consecutive waves share n-tile -> B reuse
    const int nt   = tile / Mtiles;
    const int m0   = mt * 16;
    const int n0   = nt * 16;
    const int half = lane >> 4;              // 0 or 1
    const int l15  = lane & 15;
    const int col  = n0 + l15;
    const bool colOK = (col < Nout);
    const int colSafe = colOK ? col : 0;     // OOB lanes feed cols >= Nout: never stored

    const float* a1p = Aagg  + (size_t)(m0 + l15) * F_DIM + 2 * half;
    const float* a2p = Aroot + (size_t)(m0 + l15) * F_DIM + 2 * half;
    const float* b1p = Wl + (size_t)(2 * half) * Nout + colSafe;
    const float* b2p = Wr + (size_t)(2 * half) * Nout + colSafe;

    v8f acc = {0.f, 0.f, 0.f, 0.f, 0.f, 0.f, 0.f, 0.f};

#pragma unroll 4
    for (int k = 0; k < F_DIM; k += 4) {
        v2f a1 = *(const v2f*)(a1p + k);
        v2f a2 = *(const v2f*)(a2p + k);
        v2f b1, b2;
        b1.x = b1p[(size_t)k * Nout];
        b1.y = b1p[(size_t)(k + 1) * Nout];
        b2.x = b2p[(size_t)k * Nout];
        b2.y = b2p[(size_t)(k + 1) * Nout];
        acc = __builtin_amdgcn_wmma_f32_16x16x4_f32(false, a1, false, b1,
                                                    (short)0, acc, false, false);
        acc = __builtin_amdgcn_wmma_f32_16x16x4_f32(false, a2, false, b2,
                                                    (short)0, acc, false, false);
    }

    if (colOK) {
        float bv = bias[col];
#pragma unroll
        for (int v = 0; v < 8; ++v) {
            int row = m0 + v + 8 * half;
            float r = acc[v] + bv;
            out[(size_t)row * Nout + col] = r > 0.f ? r : 0.f;
        }
    }
}

// --- in-place log_softmax over 40 classes: one wave per row ------------------
__global__ void sage_logsoftmax_kernel(float* __restrict__ out, int M) {
    int gid  = blockIdx.x * blockDim.x + threadIdx.x;
    int row  = gid >> 5;
    int lane = gid & 31;
    if (row >= M) return;
    float* p = out + (size_t)row * N_CLASSES;
    float v0 = p[lane];                              // lane 0..31 < 40 always
    bool  has2 = (lane + 32) < N_CLASSES;            // lanes 0..7 cover 32..39
    float v1 = has2 ? p[lane + 32] : -INFINITY;

    float m = fmaxf(v0, v1);
    for (int off = 16; off > 0; off >>= 1)
        m = fmaxf(m, __shfl_xor(m, off, 32));
    float s = expf(v0 - m) + (has2 ? expf(v1 - m) : 0.f);
    for (int off = 16; off > 0; off >>= 1)
        s += __shfl_xor(s, off, 32);
    float lse = m + logf(s);
    p[lane] = v0 - lse;
    if (has2) p[lane + 32] = v1 - lse;
}

extern "C" void kernel_launch(void* const* d_in, const int* in_sizes, int n_in,
                              void* d_out, int out_size, void* d_ws, size_t ws_size,
                              hipStream_t stream) {
    (void)in_sizes; (void)n_in; (void)out_size; (void)ws_size;

    const float* x   = (const float*)d_in[0];
    const int*   ei  = (const int*)d_in[1];
    const float* W1l = (const float*)d_in[2];
    const float* W1r = (const float*)d_in[3];
    const float* b1  = (const float*)d_in[4];
    const float* W2l = (const float*)d_in[5];
    const float* W2r = (const float*)d_in[6];
    const float* b2  = (const float*)d_in[7];
    float* out = (float*)d_out;

    const int* src = ei;             // edge_index[0]
    const int* dst = ei + N_EDGES;   // edge_index[1]

    const size_t aggBytes = (size_t)N_NODES * F_DIM * sizeof(float); // 25.6 MB (256-aligned)
    char* ws  = (char*)d_ws;
    float* agg = (float*)ws;              ws += aggBytes;
    float* h   = (float*)ws;              ws += aggBytes;
    float* deg = (float*)ws;              // 200 KB

    const int featN = N_NODES * F_DIM;               // 6.4M
    const int Mtiles = N_NODES / 16;                 // 3125 (exact)
    const int Nt1 = F_DIM / 16;                      // 8
    const int Nt2 = (N_CLASSES + 15) / 16;           // 3
    const int gemm1Blocks = (Mtiles * Nt1 + 7) / 8;  // 8 waves/block
    const int gemm2Blocks = (Mtiles * Nt2 + 7) / 8;

    // ---- layer 1 ----
    hipMemsetAsync(deg, 0, (size_t)N_NODES * sizeof(float), stream);
    hipMemsetAsync(agg, 0, aggBytes, stream);
    sage_deg_kernel<<<(N_EDGES + 255) / 256, 256, 0, stream>>>(dst, deg, N_EDGES);
    sage_scatter_kernel<<<6250, 256, 0, stream>>>(x, src, dst, agg, N_EDGES);
    sage_scale_kernel<<<(featN + 255) / 256, 256, 0, stream>>>(agg, deg, featN);
    sage_gemm_kernel<<<gemm1Blocks, 256, 0, stream>>>(agg, x, W1l, W1r, b1, h,
                                                      Mtiles, Nt1, F_DIM);

    // ---- layer 2 (reuse agg buffer, reuse deg) ----
    hipMemsetAsync(agg, 0, aggBytes, stream);
    sage_scatter_kernel<<<6250, 256, 0, stream>>>(h, src, dst, agg, N_EDGES);
    sage_scale_kernel<<<(featN + 255) / 256, 256, 0, stream>>>(agg, deg, featN);
    sage_gemm_kernel<<<gemm2Blocks, 256, 0, stream>>>(agg, h, W2l, W2r, b2, out,
                                                      Mtiles, Nt2, N_CLASSES);

    // ---- log_softmax in place on d_out ----
    sage_logsoftmax_kernel<<<(N_NODES * 32 + 255) / 256, 256, 0, stream>>>(out, N_NODES);
}